// ResidualTransformerLayer_89326729822667
// MI455X (gfx1250) — compile-verified
//
#include <hip/hip_runtime.h>
#include <hip/hip_bf16.h>

// ResidualTransformerLayer (TransformerConv-style GNN layer) for MI455X gfx1250.
// Memory/latency bound (~8.6 GFLOP vs ~0.6 GB traffic @ 23.3 TB/s), so GEMMs use
// exact-fp32 WMMA (v_wmma_f32_16x16x4_f32); LDS tiles staged with CDNA5 async
// global->LDS loads (ASYNCcnt) when available; edge phase uses L2 atomics.

#define HID    64
#define HEADS  4
#define DMODEL 256

typedef __attribute__((ext_vector_type(2))) float v2f;
typedef __attribute__((ext_vector_type(8))) float v8f;

#if defined(__gfx1250__) &&                                              \
    __has_builtin(__builtin_amdgcn_global_load_async_to_lds_b128) &&     \
    __has_builtin(__builtin_amdgcn_s_wait_asynccnt)
#define USE_ASYNC_LDS 1
#else
#define USE_ASYNC_LDS 0
#endif

#if USE_ASYNC_LDS
// builtin takes pointers to 16-byte int vectors in AS1 (global) / AS3 (LDS)
typedef int v4i_t __attribute__((__vector_size__(16)));

__device__ __forceinline__ __attribute__((address_space(1))) v4i_t*
as_global(const void* p) {
  return (__attribute__((address_space(1))) v4i_t*)(p);
}
__device__ __forceinline__ __attribute__((address_space(3))) v4i_t*
as_local(void* p) {
  return (__attribute__((address_space(3))) v4i_t*)(p);
}
#endif

__device__ __forceinline__ float wave_allreduce_sum(float v) {
  #pragma unroll
  for (int m = 16; m >= 1; m >>= 1) v += __shfl_xor(v, m, 32);
  return v;
}

// float atomic max via sign-split integer atomics (always compiles; init must be -inf)
__device__ __forceinline__ void atomic_max_float(float* addr, float val) {
  if (val >= 0.0f) atomicMax((int*)addr, __float_as_int(val));
  else             atomicMin((unsigned int*)addr, __float_as_uint(val));
}

// ---------------------------------------------------------------------------
// Kernel 1: fused q/k/v/skip GEMM.  Y[mi] = x @ W[mi]^T + b[mi],  [N,256] each.
// grid = (N/16, 16), block = 128 (4 waves).  Wave t = blockIdx.y*4+wid handles
// matrix mi = t>>4, 16-col tile ncol = t&15.  K = 64 -> 16 WMMA 16x16x4 steps.
// ---------------------------------------------------------------------------
__global__ void __launch_bounds__(128)
qkvs_gemm_kernel(const float* __restrict__ x,
                 const float* __restrict__ Wq, const float* __restrict__ bq,
                 const float* __restrict__ Wk, const float* __restrict__ bk,
                 const float* __restrict__ Wv, const float* __restrict__ bv,
                 const float* __restrict__ Wsk, const float* __restrict__ bsk,
                 float* __restrict__ Y, int nNodes) {
  __shared__ float sx[16][68];               // +4 pad -> conflict-free (64 banks)
  const int tid = threadIdx.x;
  const int mtile = blockIdx.x;

#if USE_ASYNC_LDS
  // 16x64 f32 tile = 256 x b128 chunks, direct global->LDS (no VGPR round-trip)
  #pragma unroll
  for (int i = tid; i < 256; i += 128) {
    int r = i >> 4, c4 = (i & 15) * 4;
    __builtin_amdgcn_global_load_async_to_lds_b128(
        as_global(x + (size_t)(mtile * 16 + r) * HID + c4),
        as_local(&sx[r][c4]), 0, 0);
  }
  __builtin_amdgcn_s_wait_asynccnt(0);
#else
  for (int i = tid; i < 16 * HID; i += 128) {
    int r = i >> 6, c = i & 63;
    sx[r][c] = x[(size_t)(mtile * 16 + r) * HID + c];
  }
#endif
  __syncthreads();

  const int wid  = tid >> 5;
  const int lane = tid & 31;
  const int t    = blockIdx.y * 4 + wid;     // 0..63
  const int mi   = t >> 4;                   // which weight matrix
  const int ncol = t & 15;                   // 16-col tile of 256

  const float* W; const float* B;
  if      (mi == 0) { W = Wq;  B = bq;  }
  else if (mi == 1) { W = Wk;  B = bk;  }
  else if (mi == 2) { W = Wv;  B = bv;  }
  else              { W = Wsk; B = bsk; }

  const int n     = lane & 15;               // output column within tile / A row
  const int khalf = (lane < 16) ? 0 : 2;     // K split across lane halves
  const float* wrow = W + (size_t)(ncol * 16 + n) * HID;
  const float  bias = B[ncol * 16 + n];      // C col = lane&15 for every vgpr

  v8f acc;
  #pragma unroll
  for (int r = 0; r < 8; ++r) acc[r] = bias;

  #pragma unroll
  for (int kk = 0; kk < 16; ++kk) {
    v2f a, b;
    a.x = sx[n][4 * kk + khalf];             // A[m][khalf],   m = lane&15
    a.y = sx[n][4 * kk + khalf + 1];         // A[m][khalf+1]
    b.x = wrow[4 * kk + khalf];              // B[khalf][n] = W[n][k]
    b.y = wrow[4 * kk + khalf + 1];
    acc = __builtin_amdgcn_wmma_f32_16x16x4_f32(
        false, a, false, b, (short)0, acc, false, false);
  }

  float* dstp = Y + (size_t)mi * nNodes * DMODEL;
  const int rowhi = (lane < 16) ? 0 : 8;
  #pragma unroll
  for (int r = 0; r < 8; ++r) {
    int row = mtile * 16 + r + rowhi;
    dstp[(size_t)row * DMODEL + ncol * 16 + n] = acc[r];
  }
}

// ---------------------------------------------------------------------------
// Kernel 2: init agg=0, mmax=-inf, denom=0
// ---------------------------------------------------------------------------
__global__ void init_kernel(float* __restrict__ agg, float* __restrict__ mmax,
                            float* __restrict__ denom, int n256, int n4) {
  int i = blockIdx.x * blockDim.x + threadIdx.x;
  if (i < n256) agg[i] = 0.0f;
  if (i < n4) { mmax[i] = -__builtin_inff(); denom[i] = 0.0f; }
}

// ---------------------------------------------------------------------------
// Kernel 3: per-edge attention logits + segment max.  One wave per edge.
// alpha[e][h] = dot64(q[dst],k[src]) / 8.  Also prefetch v[src] row into L2
// (global_prefetch_b8) to warm it for the scatter kernel.
// ---------------------------------------------------------------------------
__global__ void __launch_bounds__(256)
edge_logits_kernel(const float* __restrict__ Y, const int* __restrict__ src,
                   const int* __restrict__ dst, float* __restrict__ alpha,
                   float* __restrict__ mmax, int nNodes, int nEdges) {
  const int wid  = threadIdx.x >> 5;
  const int lane = threadIdx.x & 31;
  const int e = blockIdx.x * 8 + wid;
  if (e >= nEdges) return;
  const int s = src[e], d = dst[e];
  const float* q = Y + (size_t)d * DMODEL;                              // q @ dst
  const float* k = Y + (size_t)nNodes * DMODEL + (size_t)s * DMODEL;    // k @ src
  const float* vrow = Y + 2 * (size_t)nNodes * DMODEL + (size_t)s * DMODEL;
  __builtin_prefetch(&vrow[lane * 8], 0, 1);  // warm L2 for edge_scatter_kernel
  float p[HEADS];
  #pragma unroll
  for (int h = 0; h < HEADS; ++h) {
    int f0 = h * 64 + lane;
    p[h] = q[f0] * k[f0] + q[f0 + 32] * k[f0 + 32];
  }
  #pragma unroll
  for (int h = 0; h < HEADS; ++h) p[h] = wave_allreduce_sum(p[h]);
  if (lane < HEADS) {
    float a = p[lane] * 0.125f;              // 1/sqrt(64)
    alpha[(size_t)e * HEADS + lane] = a;
    atomic_max_float(&mmax[d * HEADS + lane], a);
  }
}

// ---------------------------------------------------------------------------
// Kernel 4: a = exp(alpha - m[dst]); denom[dst] += a  (one thread per edge-head)
// ---------------------------------------------------------------------------
__global__ void edge_softmax_kernel(const int* __restrict__ dst,
                                    float* __restrict__ alpha,
                                    const float* __restrict__ mmax,
                                    float* __restrict__ denom, int nEdges) {
  int t = blockIdx.x * blockDim.x + threadIdx.x;
  if (t >= nEdges * HEADS) return;
  int e = t >> 2, h = t & 3;
  int d = dst[e];
  float a = __expf(alpha[t] - mmax[d * HEADS + h]);
  alpha[t] = a;
  atomicAdd(&denom[d * HEADS + h], a);
}

// ---------------------------------------------------------------------------
// Kernel 5: agg[dst] += v[src] * (a/denom[dst])   (one wave per edge)
// ---------------------------------------------------------------------------
__global__ void __launch_bounds__(256)
edge_scatter_kernel(const float* __restrict__ Y, const int* __restrict__ src,
                    const int* __restrict__ dst, const float* __restrict__ alpha,
                    const float* __restrict__ denom, float* __restrict__ agg,
                    int nNodes, int nEdges) {
  const int wid  = threadIdx.x >> 5;
  const int lane = threadIdx.x & 31;
  const int e = blockIdx.x * 8 + wid;
  if (e >= nEdges) return;
  const int s = src[e], d = dst[e];
  const float* v = Y + 2 * (size_t)nNodes * DMODEL + (size_t)s * DMODEL;
  float att[HEADS];
  #pragma unroll
  for (int h = 0; h < HEADS; ++h)
    att[h] = alpha[(size_t)e * HEADS + h] / denom[d * HEADS + h];
  float* ao = agg + (size_t)d * DMODEL;
  #pragma unroll
  for (int j = 0; j < 8; ++j) {
    int f = lane + 32 * j;
    atomicAdd(&ao[f], v[f] * att[f >> 6]);
  }
}

// ---------------------------------------------------------------------------
// Kernel 6: beta gate + LayerNorm, in place on agg.  One wave per node.
// beta = sigmoid(out.(w1+w3) + skip.(w2-w3))
// ---------------------------------------------------------------------------
__global__ void __launch_bounds__(256)
gate_ln_kernel(const float* __restrict__ Y, float* __restrict__ agg,
               const float* __restrict__ Wbeta, const float* __restrict__ ln_g,
               const float* __restrict__ ln_b, int nNodes) {
  const int wid  = threadIdx.x >> 5;
  const int lane = threadIdx.x & 31;
  const int nidx = blockIdx.x * 8 + wid;
  if (nidx >= nNodes) return;
  const float* skip = Y + 3 * (size_t)nNodes * DMODEL + (size_t)nidx * DMODEL;
  float* o = agg + (size_t)nidx * DMODEL;
  float ov[8], sv[8];
  float dot_o = 0.0f, dot_s = 0.0f;
  #pragma unroll
  for (int j = 0; j < 8; ++j) {
    int f = lane + 32 * j;
    ov[j] = o[f];
    sv[j] = skip[f];
    float w1 = Wbeta[f], w2 = Wbeta[DMODEL + f], w3 = Wbeta[2 * DMODEL + f];
    dot_o += ov[j] * (w1 + w3);
    dot_s += sv[j] * (w2 - w3);
  }
  dot_o = wave_allreduce_sum(dot_o);
  dot_s = wave_allreduce_sum(dot_s);
  float beta = 1.0f / (1.0f + __expf(-(dot_o + dot_s)));
  float bl[8], sum = 0.0f, sumsq = 0.0f;
  #pragma unroll
  for (int j = 0; j < 8; ++j) {
    bl[j] = beta * sv[j] + (1.0f - beta) * ov[j];
    sum   += bl[j];
    sumsq += bl[j] * bl[j];
  }
  sum   = wave_allreduce_sum(sum);
  sumsq = wave_allreduce_sum(sumsq);
  float mu   = sum * (1.0f / DMODEL);
  float var  = sumsq * (1.0f / DMODEL) - mu * mu;
  float rstd = rsqrtf(var + 1e-5f);
  #pragma unroll
  for (int j = 0; j < 8; ++j) {
    int f = lane + 32 * j;
    o[f] = (bl[j] - mu) * rstd * ln_g[f] + ln_b[f];
  }
}

// ---------------------------------------------------------------------------
// Kernel 7: h = Z @ Wproj^T + bproj; out = relu(h + x).  K=256 -> 64 WMMA steps.
// grid = N/16, block = 128 (4 waves, one per 16-col tile of the 64 outputs).
// ---------------------------------------------------------------------------
__global__ void __launch_bounds__(128)
proj_kernel(const float* __restrict__ Z, const float* __restrict__ Wp,
            const float* __restrict__ bp, const float* __restrict__ x,
            float* __restrict__ out, int nNodes) {
  __shared__ float sz[16][260];              // +4 pad -> conflict-free
  const int tid = threadIdx.x;
  const int mtile = blockIdx.x;

#if USE_ASYNC_LDS
  // 16x256 f32 tile = 1024 x b128 chunks, direct global->LDS
  #pragma unroll
  for (int i = tid; i < 1024; i += 128) {
    int r = i >> 6, c4 = (i & 63) * 4;
    __builtin_amdgcn_global_load_async_to_lds_b128(
        as_global(Z + (size_t)(mtile * 16 + r) * DMODEL + c4),
        as_local(&sz[r][c4]), 0, 0);
  }
  __builtin_amdgcn_s_wait_asynccnt(0);
#else
  for (int i = tid; i < 16 * DMODEL; i += 128) {
    int r = i >> 8, c = i & 255;
    sz[r][c] = Z[(size_t)(mtile * 16 + r) * DMODEL + c];
  }
#endif
  __syncthreads();

  const int wid  = tid >> 5;                 // output 16-col tile (0..3)
  const int lane = tid & 31;
  const int n     = lane & 15;
  const int khalf = (lane < 16) ? 0 : 2;
  const float* wrow = Wp + (size_t)(wid * 16 + n) * DMODEL;
  const float  bias = bp[wid * 16 + n];

  v8f acc;
  #pragma unroll
  for (int r = 0; r < 8; ++r) acc[r] = bias;

  #pragma unroll
  for (int kk = 0; kk < 64; ++kk) {
    v2f a, b;
    a.x = sz[n][4 * kk + khalf];
    a.y = sz[n][4 * kk + khalf + 1];
    b.x = wrow[4 * kk + khalf];
    b.y = wrow[4 * kk + khalf + 1];
    acc = __builtin_amdgcn_wmma_f32_16x16x4_f32(
        false, a, false, b, (short)0, acc, false, false);
  }

  const int rowhi = (lane < 16) ? 0 : 8;
  #pragma unroll
  for (int r = 0; r < 8; ++r) {
    int row = mtile * 16 + r + rowhi;
    int col = wid * 16 + n;
    float h = acc[r] + x[(size_t)row * HID + col];
    out[(size_t)row * HID + col] = fmaxf(h, 0.0f);
  }
}

// ---------------------------------------------------------------------------
extern "C" void kernel_launch(void* const* d_in, const int* in_sizes, int n_in,
                              void* d_out, int out_size, void* d_ws, size_t ws_size,
                              hipStream_t stream) {
  const float* x     = (const float*)d_in[0];
  const int*   ei    = (const int*)d_in[1];
  const float* Wq    = (const float*)d_in[2];
  const float* bq    = (const float*)d_in[3];
  const float* Wk    = (const float*)d_in[4];
  const float* bk    = (const float*)d_in[5];
  const float* Wv    = (const float*)d_in[6];
  const float* bv    = (const float*)d_in[7];
  const float* Wsk   = (const float*)d_in[8];
  const float* bsk   = (const float*)d_in[9];
  const float* Wbeta = (const float*)d_in[10];
  const float* ln_g  = (const float*)d_in[11];
  const float* ln_b  = (const float*)d_in[12];
  const float* Wp    = (const float*)d_in[13];
  const float* bp    = (const float*)d_in[14];
  float* out = (float*)d_out;

  const int nNodes = in_sizes[0] / HID;   // 50000 (divisible by 16)
  const int nEdges = in_sizes[1] / 2;     // 800000
  const int* src = ei;                    // edge_index[0]
  const int* dst = ei + nEdges;           // edge_index[1]

  // workspace carve-out (256B aligned)
  char* ws = (char*)d_ws;
  size_t off = 0;
  auto alloc = [&](size_t bytes) {
    void* p = ws + off;
    off += bytes;
    off = (off + 255) & ~(size_t)255;
    return p;
  };
  float* Y     = (float*)alloc((size_t)4 * nNodes * DMODEL * sizeof(float)); // q,k,v,skip
  float* alpha = (float*)alloc((size_t)nEdges * HEADS * sizeof(float));
  float* mmax  = (float*)alloc((size_t)nNodes * HEADS * sizeof(float));
  float* denom = (float*)alloc((size_t)nNodes * HEADS * sizeof(float));
  float* agg   = (float*)alloc((size_t)nNodes * DMODEL * sizeof(float));     // reused as Z

  const int mtiles = nNodes / 16;         // exact: 3125

  qkvs_gemm_kernel<<<dim3(mtiles, 16), 128, 0, stream>>>(
      x, Wq, bq, Wk, bk, Wv, bv, Wsk, bsk, Y, nNodes);

  {
    int total = nNodes * DMODEL;
    init_kernel<<<(total + 255) / 256, 256, 0, stream>>>(
        agg, mmax, denom, total, nNodes * HEADS);
  }

  edge_logits_kernel<<<(nEdges + 7) / 8, 256, 0, stream>>>(
      Y, src, dst, alpha, mmax, nNodes, nEdges);

  edge_softmax_kernel<<<(nEdges * HEADS + 255) / 256, 256, 0, stream>>>(
      dst, alpha, mmax, denom, nEdges);

  edge_scatter_kernel<<<(nEdges + 7) / 8, 256, 0, stream>>>(
      Y, src, dst, alpha, denom, agg, nNodes, nEdges);

  gate_ln_kernel<<<(nNodes + 7) / 8, 256, 0, stream>>>(
      Y, agg, Wbeta, ln_g, ln_b, nNodes);

  proj_kernel<<<mtiles, 128, 0, stream>>>(agg, Wp, bp, x, out, nNodes);
}